// MultiHeadAttention_59863254171815
// MI455X (gfx1250) — compile-verified
//
#include <hip/hip_runtime.h>
#include <hip/hip_bf16.h>

typedef __attribute__((ext_vector_type(2))) float v2f;
typedef __attribute__((ext_vector_type(8))) float v8f;
typedef __attribute__((ext_vector_type(4))) unsigned int v4u;
typedef __attribute__((ext_vector_type(4))) int v4i;
typedef __attribute__((ext_vector_type(8))) int v8i;

#define B_    4
#define L_    1024
#define DM_   512
#define H_    8
#define DH_   64
#define ROWS  1028   // score row pitch: 1024 + 4 pad -> conflict-free column reads
#define KVROW 68     // K/V LDS row pitch: 64 + 4 pad (inserted by TDM pad feature)

// ---------------------------------------------------------------------------
// TDM: DMA a 64x64 fp32 tile (contiguous rows of 64 floats in global) into LDS
// with 4 DWORDs of padding inserted after every 64 DWORDs (row pitch 68).
// D# layout per CDNA5 ISA sect. 8 (group0 + group1, 2-D).
// This toolchain exposes the 6-arg builtin form:
//   (uint32x4 g0, int32x8 g1, int32x4 g2, int32x4 g3, int32x8 ext, i32 cpol)
// ---------------------------------------------------------------------------
__device__ __forceinline__ void tdm_load_64x64(unsigned lds_byte_off,
                                               const float* gsrc) {
    unsigned long long ga = (unsigned long long)(uintptr_t)gsrc;
    v4u g0;
    g0[0] = 1u;                                    // count=1 (valid descriptor)
    g0[1] = lds_byte_off;                          // lds_addr
    g0[2] = (unsigned)ga;                          // global_addr[31:0]
    g0[3] = (unsigned)((ga >> 32) & 0x01FFFFFFu)   // global_addr[56:32]
          | (2u << 30);                            // type = 2 ("image")
    v8i g1;
    g1[0] = (int)((2u << 16)      // data_size = 4 bytes
                | (1u << 20)      // pad_enable
                | (5u << 22)      // pad_interval: 1<<5 8B units = 64 DWORDs
                | (3u << 25));    // pad_amount: 4 DWORDs
    g1[1] = (int)(64u << 16);     // tensor_dim0 = 64  (bits 79:48)
    g1[2] = (int)(64u << 16);     // tensor_dim1 = 64  (bits 111:80)
    g1[3] = (int)(64u << 16);     // tile_dim0   = 64  (bits 127:112)
    g1[4] = (int)(64u);           // tile_dim1   = 64, tile_dim2 = 0
    g1[5] = 64;                   // tensor_dim0_stride = 64 (bits 207:160)
    g1[6] = 0;
    g1[7] = 0;
    v4i gz4 = {0, 0, 0, 0};               // groups 2/3 unused (2-D tensor)
    v8i gz8 = {0, 0, 0, 0, 0, 0, 0, 0};   // extension group unused
    __builtin_amdgcn_tensor_load_to_lds(g0, g1, gz4, gz4, gz8, 0);
}

// ---------------------------------------------------------------------------
// GEMM: Out = X @ W^T. 8 waves = 4(M) x 2(N); each wave owns a 16x64 tile
// (4 accumulators, A-fragment reused 4x). Block tile 64x128.
// ---------------------------------------------------------------------------
__global__ void proj_kernel(const float* __restrict__ X,
                            const float* __restrict__ W,
                            float* __restrict__ Out,
                            int head_layout) {
    const int w    = threadIdx.x >> 5;
    const int lid  = threadIdx.x & 31;
    const int half = lid >> 4;
    const int r    = lid & 15;
    const int m0 = blockIdx.x * 64 + (w >> 1) * 16;
    const int n0 = blockIdx.y * 128 + (w & 1) * 64;

    const float* xrow = X + (size_t)(m0 + r) * DM_ + 2 * half;
    const float* wrow = W + (size_t)(n0 + r) * DM_ + 2 * half;

    v8f c[4] = {};
    #pragma unroll 4
    for (int kb = 0; kb < DM_; kb += 4) {
        v2f a = *(const v2f*)(xrow + kb);
        #pragma unroll
        for (int t = 0; t < 4; ++t) {
            v2f bf = *(const v2f*)(wrow + (size_t)t * 16 * DM_ + kb);
            c[t] = __builtin_amdgcn_wmma_f32_16x16x4_f32(false, a, false, bf,
                                                         (short)0, c[t], false, false);
        }
    }

    #pragma unroll
    for (int t = 0; t < 4; ++t) {
        #pragma unroll
        for (int rr = 0; rr < 8; ++rr) {
            const int m = m0 + rr + 8 * half;
            const int n = n0 + t * 16 + r;
            const float v = c[t][rr];
            if (head_layout) {
                const int b = m >> 10, l = m & (L_ - 1);
                const int h = n >> 6,  d = n & (DH_ - 1);
                Out[((((size_t)b * H_ + h) * L_ + l) * DH_) + d] = v;
            } else {
                Out[(size_t)m * DM_ + n] = v;
            }
        }
    }
}

// ---------------------------------------------------------------------------
// Attention: one block per (b, h, 64-query tile); 8 waves = 4(qtile) x 2(half).
// K and V streamed through double-buffered LDS tiles by the TDM, overlapped
// with WMMA compute. Scores/probs live in LDS; attn probs stored coalesced.
// ---------------------------------------------------------------------------
__global__ void attn_kernel(const float* __restrict__ Qh,
                            const float* __restrict__ Kh,
                            const float* __restrict__ Vh,
                            const unsigned char* __restrict__ mask,
                            const int* __restrict__ adj,
                            const float* __restrict__ edge_bias,
                            float* __restrict__ attn_out,
                            float* __restrict__ AO) {
    extern __shared__ float smem[];
    float* sc      = smem;                      // [64][ROWS]
    float* kv      = smem + 64 * ROWS;          // 2 x [64][KVROW] TDM buffers
    float* red     = kv + 2 * 64 * KVROW;       // [64][4] softmax partials
    float* rowstat = red + 256;                 // [64] row max / inv-sum

    const unsigned kv_off = (unsigned)(64u * ROWS * sizeof(float)); // kv byte off
    const unsigned buf_bytes = (unsigned)(64u * KVROW * sizeof(float));

    const int q0 = blockIdx.x * 64;
    const int h  = blockIdx.y;
    const int b  = blockIdx.z;
    const int tid  = threadIdx.x;
    const int w    = tid >> 5;
    const int lid  = tid & 31;
    const int half = lid >> 4;
    const int r    = lid & 15;
    const int wq   = w >> 1;      // 0..3 : q-tile
    const int wk   = w & 1;       // 0..1 : key half (stage A) / d half (stage C)

    const float eb0 = edge_bias[h * 2 + 0];
    const float eb1 = edge_bias[h * 2 + 1];
    const size_t bh = ((size_t)b * H_ + h) * L_;
    const float* Ksrc = Kh + bh * DH_;
    const float* Vsrc = Vh + bh * DH_;

    // Cache this wave's Q tile (16 q x 64 d) as 16 A-fragments.
    v2f qf[16];
    {
        const float* qrow = Qh + (bh + q0 + wq * 16 + r) * DH_ + 2 * half;
        #pragma unroll
        for (int i = 0; i < 16; ++i) qf[i] = *(const v2f*)(qrow + 4 * i);
    }

    // ---- Stage A: scores = QK^T/8 + edge_bias, masked -> LDS ----
    if (w == 0) tdm_load_64x64(kv_off, Ksrc);
    for (int c = 0; c < 16; ++c) {
        if (w == 0) {
            if (c < 15) {
                tdm_load_64x64(kv_off + ((c + 1) & 1) * buf_bytes, Ksrc + (c + 1) * 4096);
                __builtin_amdgcn_s_wait_tensorcnt(1);   // chunk c landed
            } else {
                __builtin_amdgcn_s_wait_tensorcnt(0);
            }
        }
        __syncthreads();
        const float* kbuf = kv + (c & 1) * (64 * KVROW);
        #pragma unroll
        for (int nt = 0; nt < 2; ++nt) {
            const int koff = wk * 32 + nt * 16;
            const float* krow = kbuf + (koff + r) * KVROW + 2 * half;
            v8f acc = {};
            #pragma unroll
            for (int i = 0; i < 16; ++i) {
                v2f bk = *(const v2f*)(krow + 4 * i);
                acc = __builtin_amdgcn_wmma_f32_16x16x4_f32(false, qf[i], false, bk,
                                                            (short)0, acc, false, false);
            }
            const int key = c * 64 + koff + r;
            #pragma unroll
            for (int rr = 0; rr < 8; ++rr) {
                const int q = wq * 16 + rr + 8 * half;
                const size_t mi = ((size_t)b * L_ + (q0 + q)) * L_ + key;
                float v = acc[rr] * 0.125f;              // 1/sqrt(64)
                v += (adj[mi] == 0) ? eb0 : eb1;
                if (!mask[mi]) v = -1e9f;
                sc[q * ROWS + key] = v;
            }
        }
        __syncthreads();
    }

    // ---- Stage B: softmax (4 threads per row) ----
    {
        const int row = tid >> 2, cc = tid & 3;
        float* prow = sc + row * ROWS;
        const int k0 = cc * 256;

        float m = -1e30f;
        for (int k = k0; k < k0 + 256; ++k) m = fmaxf(m, prow[k]);
        red[row * 4 + cc] = m;
        __syncthreads();
        if (cc == 0)
            rowstat[row] = fmaxf(fmaxf(red[row * 4 + 0], red[row * 4 + 1]),
                                 fmaxf(red[row * 4 + 2], red[row * 4 + 3]));
        __syncthreads();
        const float mm = rowstat[row];
        float s = 0.0f;
        for (int k = k0; k < k0 + 256; ++k) {
            const float e = __expf(prow[k] - mm);
            prow[k] = e;
            s += e;
        }
        red[row * 4 + cc] = s;
        __syncthreads();
        if (cc == 0)
            rowstat[row] = 1.0f / (red[row * 4 + 0] + red[row * 4 + 1] +
                                   red[row * 4 + 2] + red[row * 4 + 3]);
        __syncthreads();
    }

    // Normalize + write attn probs (coalesced: consecutive lanes -> keys).
    {
        float* gbase = attn_out + (((size_t)b * H_ + h) * L_ + q0) * L_;
        for (int idx = tid; idx < 64 * 1024; idx += 256) {
            const int row = idx >> 10, key = idx & (L_ - 1);
            const float p = sc[row * ROWS + key] * rowstat[row];
            sc[row * ROWS + key] = p;
            gbase[(size_t)row * L_ + key] = p;
        }
    }
    __syncthreads();

    // ---- Stage C: out(64q x 64d) = P @ V ; wave owns 16q x 32d ----
    if (w == 0) tdm_load_64x64(kv_off, Vsrc);
    {
        const int dbase = wk * 32;
        v8f c0 = {}, c1 = {};
        for (int c = 0; c < 16; ++c) {
            if (w == 0) {
                if (c < 15) {
                    tdm_load_64x64(kv_off + ((c + 1) & 1) * buf_bytes, Vsrc + (c + 1) * 4096);
                    __builtin_amdgcn_s_wait_tensorcnt(1);
                } else {
                    __builtin_amdgcn_s_wait_tensorcnt(0);
                }
            }
            __syncthreads();
            const float* vbuf = kv + (c & 1) * (64 * KVROW);
            const float* arow = sc + (wq * 16 + r) * ROWS + c * 64 + 2 * half;
            #pragma unroll
            for (int s = 0; s < 16; ++s) {
                v2f a = *(const v2f*)(arow + 4 * s);               // P[q][k..k+1]
                const float* vrow = vbuf + (4 * s + 2 * half) * KVROW + dbase + r;
                v2f b0, b1;
                b0.x = vrow[0];       b0.y = vrow[KVROW];          // V[k][d], V[k+1][d]
                b1.x = vrow[16];      b1.y = vrow[KVROW + 16];
                c0 = __builtin_amdgcn_wmma_f32_16x16x4_f32(false, a, false, b0,
                                                           (short)0, c0, false, false);
                c1 = __builtin_amdgcn_wmma_f32_16x16x4_f32(false, a, false, b1,
                                                           (short)0, c1, false, false);
            }
            __syncthreads();
        }
        #pragma unroll
        for (int rr = 0; rr < 8; ++rr) {
            const int q = q0 + wq * 16 + rr + 8 * half;
            float* o = AO + ((size_t)b * L_ + q) * DM_ + h * DH_ + dbase + r;
            o[0]  = c0[rr];     // merged-heads layout [B, L, H*DH]
            o[16] = c1[rr];
        }
    }
}

// ---------------------------------------------------------------------------
extern "C" void kernel_launch(void* const* d_in, const int* in_sizes, int n_in,
                              void* d_out, int out_size, void* d_ws, size_t ws_size,
                              hipStream_t stream) {
    const float*         queries = (const float*)d_in[0];
    const float*         keys    = (const float*)d_in[1];
    const float*         values  = (const float*)d_in[2];
    const unsigned char* mask    = (const unsigned char*)d_in[3]; // jnp.bool_
    const int*           adj     = (const int*)d_in[4];
    const float*         Wq      = (const float*)d_in[5];
    const float*         Wk      = (const float*)d_in[6];
    const float*         Wv      = (const float*)d_in[7];
    const float*         Wo      = (const float*)d_in[8];
    const float*         ebias   = (const float*)d_in[9];

    float* out  = (float*)d_out;                        // [B, L, 512]
    float* attn = out + (size_t)B_ * L_ * DM_;          // [B, H, L, L]

    const size_t N_PROJ = (size_t)B_ * L_ * DM_;        // 2M floats
    float* ws = (float*)d_ws;
    float* Qh = ws;                                     // [B,H,L,64]
    float* Kh = ws + 1 * N_PROJ;
    float* Vh = ws + 2 * N_PROJ;
    float* AO = ws + 3 * N_PROJ;                        // [B,L,512] merged heads

    const dim3 gproj((B_ * L_) / 64, DM_ / 128);        // 64 x 4 blocks
    proj_kernel<<<gproj, 256, 0, stream>>>(queries, Wq, Qh, 1);
    proj_kernel<<<gproj, 256, 0, stream>>>(keys,    Wk, Kh, 1);
    proj_kernel<<<gproj, 256, 0, stream>>>(values,  Wv, Vh, 1);

    const size_t smem = (size_t)(64 * ROWS + 2 * 64 * KVROW + 256 + 64) * sizeof(float);
    attn_kernel<<<dim3(L_ / 64, H_, B_), 256, smem, stream>>>(
        Qh, Kh, Vh, mask, adj, ebias, attn, AO);

    proj_kernel<<<gproj, 256, 0, stream>>>(AO, Wo, out, 0);
}